// MultiScaleDeformableAttention_45930380263466
// MI455X (gfx1250) — compile-verified
//
#include <hip/hip_runtime.h>
#include <hip/hip_bf16.h>
#include <stdint.h>

// Multi-Scale Deformable Attention forward, MI455X (gfx1250, wave32).
// Mapping: 1 block = 1 query; 8 waves = 8 heads; lane = channel (D=32).
// Every corner gather is one coalesced 128B transaction per wave.
// Metadata (sampling locs + attention weights, 1.5KB/query) is staged to LDS
// with CDNA5 async global->LDS loads (ASYNCcnt) when the builtin exists.

#define MSDA_HEADS 8
#define MSDA_D     32
#define MSDA_L     4
#define MSDA_P     4
#define MSDA_HD    (MSDA_HEADS * MSDA_D)               // 256 = block size
#define MSDA_LOCQ  (MSDA_HEADS * MSDA_L * MSDA_P * 2)  // 256 floats / query
#define MSDA_AWQ   (MSDA_HEADS * MSDA_L * MSDA_P)      // 128 floats / query

#if defined(__gfx1250__) && __has_builtin(__builtin_amdgcn_global_load_async_to_lds_b32)
#define MSDA_ASYNC 1
#else
#define MSDA_ASYNC 0
#endif

__global__ __launch_bounds__(MSDA_HD) void msda_fwd_kernel(
    const float* __restrict__ value,          // [n_tokens, heads, D]
    const long long* __restrict__ spatial_shapes, // [L][2] (H,W)
    const long long* __restrict__ level_start,    // [L]
    const float* __restrict__ loc,            // [Q, heads, L, P, 2]
    const float* __restrict__ aw,             // [Q, heads, L, P]
    float* __restrict__ out,                  // [Q, heads*D]
    int Q)
{
    __shared__ float s_loc[MSDA_LOCQ];
    __shared__ float s_aw[MSDA_AWQ];

    const int q   = blockIdx.x;
    const int tid = threadIdx.x;           // tid = head*32 + channel
    if (q >= Q) return;

    const float* locq = loc + (size_t)q * MSDA_LOCQ;
    const float* awq  = aw  + (size_t)q * MSDA_AWQ;

    // Speculative prefetch of next query's metadata (global_prefetch_b8;
    // OOB speculative prefetches are silently dropped per CDNA5 ISA).
    __builtin_prefetch(locq + MSDA_LOCQ, 0, 1);

#if MSDA_ASYNC
    // CDNA5 async global->LDS copy, tracked by ASYNCcnt.
    // Builtin signature (from hipcc diagnostics): (as(1) int*, as(3) int*, imm, imm)
    __builtin_amdgcn_global_load_async_to_lds_b32(
        (__attribute__((address_space(1))) int*)(locq + tid),
        (__attribute__((address_space(3))) int*)(&s_loc[tid]), 0, 0);
    if (tid < MSDA_AWQ) {
        __builtin_amdgcn_global_load_async_to_lds_b32(
            (__attribute__((address_space(1))) int*)(awq + tid),
            (__attribute__((address_space(3))) int*)(&s_aw[tid]), 0, 0);
    }
#if __has_builtin(__builtin_amdgcn_s_wait_asynccnt)
    __builtin_amdgcn_s_wait_asynccnt(0);
#else
    asm volatile("s_wait_asynccnt 0" ::: "memory");
#endif
#else
    s_loc[tid] = locq[tid];
    if (tid < MSDA_AWQ) s_aw[tid] = awq[tid];
#endif
    __syncthreads();

    const int h = tid >> 5;                                   // head id
    const float* lph = s_loc + h * (MSDA_L * MSDA_P * 2);
    const float* wph = s_aw  + h * (MSDA_L * MSDA_P);

    float acc = 0.0f;

#pragma unroll
    for (int l = 0; l < MSDA_L; ++l) {
        const int H  = (int)spatial_shapes[2 * l + 0];
        const int W  = (int)spatial_shapes[2 * l + 1];
        const int ls = (int)level_start[l];
        const float Hf = (float)H, Wf = (float)W;

#pragma unroll
        for (int p = 0; p < MSDA_P; ++p) {
            const float xn  = lph[l * (MSDA_P * 2) + p * 2 + 0];
            const float yn  = lph[l * (MSDA_P * 2) + p * 2 + 1];
            const float wgt = wph[l * MSDA_P + p];

            // align_corners=False grid sample with zero padding
            const float x   = xn * Wf - 0.5f;
            const float y   = yn * Hf - 0.5f;
            const float x0f = floorf(x);
            const float y0f = floorf(y);
            const int   x0  = (int)x0f;
            const int   y0  = (int)y0f;
            const float fx  = x - x0f;
            const float fy  = y - y0f;
            const int   x1  = x0 + 1;
            const int   y1  = y0 + 1;

            const bool vx0 = (unsigned)x0 < (unsigned)W;
            const bool vx1 = (unsigned)x1 < (unsigned)W;
            const bool vy0 = (unsigned)y0 < (unsigned)H;
            const bool vy1 = (unsigned)y1 < (unsigned)H;

            // value index = token*256 + head*32 + d = token*256 + tid
            const int r0 = (ls + y0 * W) * MSDA_HD + tid;
            const int r1 = (ls + y1 * W) * MSDA_HD + tid;

            const float g00 = (vx0 && vy0) ? value[r0 + x0 * MSDA_HD] : 0.0f;
            const float g01 = (vx1 && vy0) ? value[r0 + x1 * MSDA_HD] : 0.0f;
            const float g10 = (vx0 && vy1) ? value[r1 + x0 * MSDA_HD] : 0.0f;
            const float g11 = (vx1 && vy1) ? value[r1 + x1 * MSDA_HD] : 0.0f;

            const float gx0 = 1.0f - fx;
            const float gy0 = 1.0f - fy;
            const float s = fmaf(gx0 * gy0, g00,
                            fmaf(fx  * gy0, g01,
                            fmaf(gx0 * fy , g10,
                                 fx  * fy * g11)));
            acc = fmaf(wgt, s, acc);
        }
    }

    out[(size_t)q * MSDA_HD + tid] = acc;  // one coalesced 1KB store / block
}

extern "C" void kernel_launch(void* const* d_in, const int* in_sizes, int n_in,
                              void* d_out, int out_size, void* d_ws, size_t ws_size,
                              hipStream_t stream) {
    const float*     value  = (const float*)d_in[0];
    const long long* shapes = (const long long*)d_in[1];
    const long long* lstart = (const long long*)d_in[2];
    const float*     loc    = (const float*)d_in[3];
    const float*     aw     = (const float*)d_in[4];
    // d_in[5] = im2col_step (unused)
    float* out = (float*)d_out;

    const int Q = out_size / MSDA_HD;  // out is [Q, heads*D]

    dim3 grid(Q);
    dim3 block(MSDA_HD);
    hipLaunchKernelGGL(msda_fwd_kernel, grid, block, 0, stream,
                       value, shapes, lstart, loc, aw, out, Q);
}